// CapsGNN_47648367182455
// MI455X (gfx1250) — compile-verified
//
#include <hip/hip_runtime.h>

// ---------------------------------------------------------------------------
// CapsGNN forward for MI455X (gfx1250, wave32).
// All matrix math uses V_WMMA_F32_16X16X4_F32 (f32-faithful to reference).
// GEMMs are templated on strides/K so the hot loop is branch-free with
// immediate-offset addressing; attention weight is zero-padded to Nc=128 so
// no bounds guards (and no EXEC games) are needed anywhere near the WMMAs.
// ---------------------------------------------------------------------------

typedef __attribute__((ext_vector_type(2))) float v2f;
typedef __attribute__((ext_vector_type(8))) float v8f;

static __device__ __forceinline__ v8f wmma4(v2f a, v2f b, v8f c) {
  // (neg_a, A, neg_b, B, c_mod, C, reuse_a, reuse_b)
  return __builtin_amdgcn_wmma_f32_16x16x4_f32(false, a, false, b, (short)0, c,
                                               false, false);
}

// ---- dims -----------------------------------------------------------------
// B=8 N=512 F=128 D_CAP=16 L=4 H=16 K=8 OUT=16 GDIM=256 HID=68 NL=2048 BN=4096

// ---------------------------------------------------------------------------
// enc_W [c=16, f=128, d=16]  ->  encWt [f=128, g=c*16+d (256)]
__global__ void k_transpose_encw(const float* __restrict__ w,
                                 float* __restrict__ wt) {
  int idx = blockIdx.x * 256 + threadIdx.x;  // 32768
  int d = idx & 15, f = (idx >> 4) & 127, c = idx >> 11;
  wt[f * 256 + c * 16 + d] = w[idx];
}

// attn_W1 [1024, 68] -> zero-padded [1024, 128]
__global__ void k_padw1(const float* __restrict__ w1, float* __restrict__ w1p) {
  int idx = blockIdx.x * 256 + threadIdx.x;  // 131072
  int col = idx & 127, row = idx >> 7;
  w1p[idx] = (col < 68) ? w1[row * 68 + col] : 0.f;
}

// deg[b,i] = rsqrt(max(rowsum(adj_loop), 1)),  adj_loop diag forced to 1
__global__ void k_deg(const float* __restrict__ adj, float* __restrict__ deg) {
  int lane = threadIdx.x & 31, wave = threadIdx.x >> 5;
  int r = blockIdx.x * 8 + wave;  // 0..4095
  int b = r >> 9, i = r & 511;
  const float* row = adj + ((long)b * 512 + i) * 512;
  float s = 0.f;
  for (int j = lane; j < 512; j += 32) s += row[j];
#pragma unroll
  for (int o = 16; o > 0; o >>= 1) s += __shfl_xor(s, o);
  if (lane == 0) {
    float loopsum = s - row[i] + 1.0f;  // diag -> 1
    deg[r] = rsqrtf(fmaxf(loopsum, 1.0f));
  }
}

// adj_n[b,i,j] = deg_i * adj_loop[i,j] * deg_j
__global__ void k_adjn(const float* __restrict__ adj,
                       const float* __restrict__ deg,
                       float* __restrict__ adjn) {
  long idx = (long)blockIdx.x * 256 + threadIdx.x;  // 2,097,152
  int j = (int)(idx & 511);
  int i = (int)((idx >> 9) & 511);
  int b = (int)(idx >> 18);
  float v = (i == j) ? 1.0f : adj[idx];
  adjn[idx] = v * deg[b * 512 + i] * deg[b * 512 + j];
}

// ---------------------------------------------------------------------------
// Batched WMMA f32 GEMM: C[z] = epilogue( A[z](MxK) @ B[z](KxNc) ).
// 256 thr = 8 waves; wave -> 16x64 strip (4 accumulators, A reused 4x);
// block tile 128x64.  Requires M % 128 == 0 and Nc % 64 == 0 (grid-exact).
// MODE: 0=none  1=relu(x+bias)*mask  2=tanh((x+bias)*mask)  3=tanh(x)
template <int LDA, int LDB, int LDC, int K, int MODE>
__global__ void k_gemm(const float* __restrict__ A, long sA,
                       const float* __restrict__ Bm, long sB,
                       float* __restrict__ C, long sC,
                       const float* __restrict__ bias,
                       const float* __restrict__ mask, int maskStride) {
  int z = blockIdx.z;
  A += (long)z * sA;
  Bm += (long)z * sB;
  C += (long)z * sC;
  int lane = threadIdx.x & 31, wave = threadIdx.x >> 5;
  int half = lane >> 4, lm = lane & 15;
  int row0 = blockIdx.y * 128 + wave * 16;
  int col0 = blockIdx.x * 64;

  v8f acc[4];
#pragma unroll
  for (int t = 0; t < 4; t++)
    acc[t] = (v8f){0.f, 0.f, 0.f, 0.f, 0.f, 0.f, 0.f, 0.f};

  const float* arow = A + (long)(row0 + lm) * LDA + 2 * half;  // a.x/a.y pair
  const float* brow = Bm + (long)(2 * half) * LDB + col0 + lm;

  for (int kk = 0; kk < K; kk += 32) {
    if (kk + 32 < K) {  // uniform branch: prefetch next chunk
      __builtin_prefetch(arow + kk + 32, 0, 1);
      __builtin_prefetch(brow + (long)(kk + 32) * LDB, 0, 1);
    }
#pragma unroll
    for (int t8 = 0; t8 < 8; t8++) {
      int k0 = kk + t8 * 4;
      v2f a;
      a.x = arow[k0];
      a.y = arow[k0 + 1];
      const float* bp = brow + (long)k0 * LDB;
#pragma unroll
      for (int t = 0; t < 4; t++) {
        v2f b;
        b.x = bp[t * 16];
        b.y = bp[LDB + t * 16];
        acc[t] = wmma4(a, b, acc[t]);
      }
    }
  }

#pragma unroll
  for (int v = 0; v < 8; v++) {
    int r = row0 + v + 8 * half;
    float mk = 1.0f;
    if constexpr (MODE == 1 || MODE == 2) mk = mask[z * maskStride + r];
#pragma unroll
    for (int t = 0; t < 4; t++) {
      int c = col0 + t * 16 + lm;
      float xv = acc[t][v];
      if constexpr (MODE == 1) xv = fmaxf(xv + bias[c], 0.f) * mk;
      else if constexpr (MODE == 2) xv = tanhf((xv + bias[c]) * mk);
      else if constexpr (MODE == 3) xv = tanhf(xv);
      C[(long)r * LDC + c] = xv;
    }
  }
}

// ---------------------------------------------------------------------------
// squash along a 256-wide row (eps=1e-11). Optional scale (a_j) output.
__global__ void k_squash256(const float* __restrict__ in,
                            float* __restrict__ out,
                            float* __restrict__ scale_out) {
  __shared__ float red[256];
  int row = blockIdx.x, t = threadIdx.x;
  float x = in[(long)row * 256 + t];
  red[t] = x * x;
  __syncthreads();
#pragma unroll
  for (int s = 128; s > 0; s >>= 1) {
    if (t < s) red[t] += red[t + s];
    __syncthreads();
  }
  float sq = red[0];
  float scale = sq / (1.0f + sq);
  float inv = 1.0f / sqrtf(sq + 1e-11f);
  out[(long)row * 256 + t] = scale * x * inv;
  if (scale_out != nullptr && t == 0) scale_out[row] = scale;
}

// ---------------------------------------------------------------------------
// attention stage 2: logits = hact(4096x128pad) @ W2(68x4); mask-fill 0;
// softmax over L=4; scale hidden row (1024) in place. One wave per row.
__global__ void k_attn2(const float* __restrict__ hact,
                        const float* __restrict__ W2,
                        const float* __restrict__ mask,
                        float* __restrict__ hidden) {
  int lane = threadIdx.x & 31, wave = threadIdx.x >> 5;
  int m = blockIdx.x * 8 + wave;  // 0..4095
  float l0 = 0.f, l1 = 0.f, l2 = 0.f, l3 = 0.f;
  for (int h = lane; h < 68; h += 32) {
    float x = hact[(long)m * 128 + h];
    l0 += x * W2[h * 4 + 0];
    l1 += x * W2[h * 4 + 1];
    l2 += x * W2[h * 4 + 2];
    l3 += x * W2[h * 4 + 3];
  }
#pragma unroll
  for (int o = 16; o > 0; o >>= 1) {
    l0 += __shfl_xor(l0, o);
    l1 += __shfl_xor(l1, o);
    l2 += __shfl_xor(l2, o);
    l3 += __shfl_xor(l3, o);
  }
  if (mask[m] == 0.f) { l0 = l1 = l2 = l3 = 0.f; }  // masked_fill(0)
  float mx = fmaxf(fmaxf(l0, l1), fmaxf(l2, l3));
  float e0 = expf(l0 - mx), e1 = expf(l1 - mx), e2 = expf(l2 - mx),
        e3 = expf(l3 - mx);
  float inv = 1.0f / (e0 + e1 + e2 + e3);
  float a0 = e0 * inv, a1 = e1 * inv, a2 = e2 * inv, a3 = e3 * inv;
  for (int g = lane; g < 1024; g += 32) {
    int l = g >> 8;
    float a = (l == 0) ? a0 : (l == 1) ? a1 : (l == 2) ? a2 : a3;
    hidden[(long)m * 1024 + g] *= a;
  }
}

// ---------------------------------------------------------------------------
// tmp2[j, m, h, o] = sum_k sec_W[k, m, o] * sec_alpha[j, h, k]
__global__ void k_tmp2(const float* __restrict__ secW,
                       const float* __restrict__ alpha,
                       float* __restrict__ tmp2) {
  long idx = (long)blockIdx.x * 256 + threadIdx.x;  // 8,388,608
  int o = (int)(idx & 15);
  int h = (int)((idx >> 4) & 15);
  int m = (int)((idx >> 8) & 15);
  long j = idx >> 12;
  float s = 0.f;
#pragma unroll
  for (int k = 0; k < 8; k++)
    s += secW[k * 256 + m * 16 + o] * alpha[(j * 16 + h) * 8 + k];
  tmp2[idx] = s;
}

// ---------------------------------------------------------------------------
// u[b,j,h,(i,o)] = mask[b, j%512] * ( first[b,j](16x16) @ tmp2[j](16x256) ).
// One wave per (b,j); A fragments loaded once, 16 h-tiles x 4 WMMA each.
__global__ void k_u(const float* __restrict__ hidden,
                    const float* __restrict__ tmp2,
                    const float* __restrict__ mask, float* __restrict__ u) {
  int lane = threadIdx.x & 31, wave = threadIdx.x >> 5;
  int wid = blockIdx.x * 8 + wave;  // 0..16383
  int b = wid >> 11, j = wid & 2047;
  int half = lane >> 4, lm = lane & 15;
  const float* A = hidden + (long)b * 524288 + (long)j * 256;  // [i][m]
  v2f a[4];
#pragma unroll
  for (int t = 0; t < 4; t++) {
    a[t].x = A[lm * 16 + t * 4 + 2 * half];
    a[t].y = A[lm * 16 + t * 4 + 2 * half + 1];
  }
  const float* Bm = tmp2 + (long)j * 4096;  // rows m, ld 256
  float scale = mask[b * 512 + (j & 511)];  // torch mask.repeat quirk
  for (int h = 0; h < 16; h++) {
    v8f acc = {0.f, 0.f, 0.f, 0.f, 0.f, 0.f, 0.f, 0.f};
#pragma unroll
    for (int t = 0; t < 4; t++) {
      int kb = t * 4 + 2 * half;
      v2f bb;
      bb.x = Bm[kb * 256 + h * 16 + lm];
      bb.y = Bm[(kb + 1) * 256 + h * 16 + lm];
      acc = wmma4(a[t], bb, acc);
    }
    float* up = u + (((long)(b * 2048 + j) * 16 + h) << 8);
#pragma unroll
    for (int v = 0; v < 8; v++)
      up[(v + 8 * half) * 16 + lm] = acc[v] * scale;  // i = v+8*half, o = lm
  }
}

// ---------------------------------------------------------------------------
// c[b,j,:] = softmax_h( bc[b,j,:] * (use_mask ? mask[b, j%512] : 1) )
__global__ void k_csoft(const float* __restrict__ bc,
                        const float* __restrict__ mask, float* __restrict__ cc,
                        int use_mask) {
  int idx = blockIdx.x * 256 + threadIdx.x;  // 16384 = b*2048+j
  int b = idx >> 11, j = idx & 2047;
  float mk = use_mask ? mask[b * 512 + (j & 511)] : 1.0f;
  float v[16];
  float mx = -1e30f;
#pragma unroll
  for (int h = 0; h < 16; h++) {
    v[h] = bc[(long)idx * 16 + h] * mk;
    mx = fmaxf(mx, v[h]);
  }
  float s = 0.f;
#pragma unroll
  for (int h = 0; h < 16; h++) {
    v[h] = expf(v[h] - mx);
    s += v[h];
  }
  float inv = 1.0f / s;
#pragma unroll
  for (int h = 0; h < 16; h++) cc[(long)idx * 16 + h] = v[h] * inv;
}

// s[b,h,d] = sum_j c[b,j,h] * u[b,j,h,d].  Block per (b,h), thread per d.
__global__ void k_sred(const float* __restrict__ cc, const float* __restrict__ u,
                       float* __restrict__ s) {
  int bh = blockIdx.x;  // 0..127
  int b = bh >> 4, h = bh & 15;
  int d = threadIdx.x;
  float acc = 0.f;
  for (int j = 0; j < 2048; j++) {
    long row = (long)(b * 2048 + j) * 16 + h;
    acc += cc[row] * u[(row << 8) + d];
  }
  s[(long)bh * 256 + d] = acc;
}

// bc[b,j,h] += sum_d u[b,j,h,d] * v[b,h,d].  One wave per (b,j,h).
__global__ void k_bupd(const float* __restrict__ u, const float* __restrict__ v,
                       float* __restrict__ bc) {
  int lane = threadIdx.x & 31, wave = threadIdx.x >> 5;
  long wid = (long)blockIdx.x * 8 + wave;  // 0..262143 = (b*2048+j)*16+h
  int b = (int)(wid >> 15), h = (int)(wid & 15);
  const float* up = u + (wid << 8);
  const float* vp = v + ((long)(b * 16 + h) << 8);
  float dot = 0.f;
  for (int d = lane; d < 256; d += 32) dot += up[d] * vp[d];
#pragma unroll
  for (int o = 16; o > 0; o >>= 1) dot += __shfl_xor(dot, o);
  if (lane == 0) bc[wid] += dot;
}

// ---------------------------------------------------------------------------
extern "C" void kernel_launch(void* const* d_in, const int* in_sizes, int n_in,
                              void* d_out, int out_size, void* d_ws,
                              size_t ws_size, hipStream_t stream) {
  const float* x = (const float*)d_in[0];
  const float* adj = (const float*)d_in[1];
  const float* mask = (const float*)d_in[2];
  const float* enc_W = (const float*)d_in[3];
  const float* enc_b = (const float*)d_in[4];
  const float* gcn_W = (const float*)d_in[5];
  const float* gcn_b = (const float*)d_in[6];
  const float* attn_W1 = (const float*)d_in[7];
  const float* attn_W2 = (const float*)d_in[8];
  const float* sec_W = (const float*)d_in[9];
  const float* sec_alpha = (const float*)d_in[10];
  (void)in_sizes; (void)n_in; (void)out_size; (void)ws_size;

  float* ws = (float*)d_ws;
  float* adjn = ws;                    // 2,097,152
  float* deg = adjn + 2097152;         // 4,096
  float* encWt = deg + 4096;           // 32,768
  float* w1pad = encWt + 32768;        // 131,072 (1024x128)
  float* h0 = w1pad + 131072;          // 1,048,576
  float* hw = h0 + 1048576;            // 1,048,576
  float* hidden = hw + 1048576;        // 4,194,304  [B,N,L,256]
  float* hact = hidden + 4194304;      // 524,288 (4096x128 padded)
  float* tmp2 = hact + 524288;         // 8,388,608  [j,m,h,o]
  float* u = tmp2 + 8388608;           // 67,108,864 [b,j,h,256]
  float* bc = u + 67108864;            // 262,144
  float* cc = bc + 262144;             // 262,144
  float* sbuf = cc + 262144;           // 32,768
  float* vbuf = sbuf + 32768;          // 32,768

  dim3 blk(256);

  // --- firstCapsuleLayer ---
  k_transpose_encw<<<128, blk, 0, stream>>>(enc_W, encWt);
  k_padw1<<<512, blk, 0, stream>>>(attn_W1, w1pad);
  k_deg<<<512, blk, 0, stream>>>(adj, deg);
  k_adjn<<<8192, blk, 0, stream>>>(adj, deg, adjn);

  // h0 = squash( relu(x @ encWt + enc_b) * mask )   (4096x128)@(128x256)
  k_gemm<128, 256, 256, 128, 1>
      <<<dim3(4, 32, 1), blk, 0, stream>>>(x, 0, encWt, 0, h0, 0, enc_b, mask, 0);
  k_squash256<<<4096, blk, 0, stream>>>(h0, h0, nullptr);

  // GCN layers: hidden[:,:,l,:] = tanh((adj_n @ (h @ W_l) + b_l) * mask)
  for (int l = 0; l < 4; l++) {
    if (l == 0) {
      k_gemm<256, 256, 256, 256, 0><<<dim3(4, 32, 1), blk, 0, stream>>>(
          h0, 0, gcn_W, 0, hw, 0, nullptr, nullptr, 0);
    } else {
      k_gemm<1024, 256, 256, 256, 0><<<dim3(4, 32, 1), blk, 0, stream>>>(
          hidden + (l - 1) * 256, 0, gcn_W + l * 65536, 0, hw, 0, nullptr,
          nullptr, 0);
    }
    k_gemm<512, 256, 1024, 512, 2><<<dim3(4, 4, 8), blk, 0, stream>>>(
        adjn, (long)512 * 512, hw, (long)512 * 256, hidden + l * 256,
        (long)512 * 1024, gcn_b + l * 256, mask, 512);
  }

  // attention: hact = tanh(hidden_flat @ W1pad); softmax(W2 logits); scale
  k_gemm<1024, 128, 128, 1024, 3><<<dim3(2, 32, 1), blk, 0, stream>>>(
      hidden, 0, w1pad, 0, hact, 0, nullptr, nullptr, 0);
  k_attn2<<<512, blk, 0, stream>>>(hact, attn_W2, mask, hidden);

  // --- SecondaryCapsuleLayer ---
  k_tmp2<<<32768, blk, 0, stream>>>(sec_W, sec_alpha, tmp2);
  k_u<<<2048, blk, 0, stream>>>(hidden, tmp2, mask, u);

  hipMemsetAsync(bc, 0, 262144 * sizeof(float), stream);
  for (int it = 0; it < 2; it++) {  // NUM_ITER-1 routing iterations
    k_csoft<<<64, blk, 0, stream>>>(bc, mask, cc, 0);
    k_sred<<<128, blk, 0, stream>>>(cc, u, sbuf);
    k_squash256<<<128, blk, 0, stream>>>(sbuf, vbuf, nullptr);
    k_bupd<<<32768, blk, 0, stream>>>(u, vbuf, bc);
  }
  // final pass: b *= mask2, c = softmax, s, (v, a_j) -> d_out
  k_csoft<<<64, blk, 0, stream>>>(bc, mask, cc, 1);
  k_sred<<<128, blk, 0, stream>>>(cc, u, sbuf);
  float* outv = (float*)d_out;                         // v: 32768 floats
  k_squash256<<<128, blk, 0, stream>>>(sbuf, outv, outv + 32768);  // a_j: 128
}